// GatMNN_76364518523063
// MI455X (gfx1250) — compile-verified
//
#include <hip/hip_runtime.h>
#include <math.h>

#define N_NODES 50000
#define N_EDGES 800000
#define IN_FEATS 9
#define IN_FEATS_PAD 12
#define H_FEATS 64
#define HEADS 3
#define N_CLASSES 2
#define NEG_SLOPE 0.2f

typedef __attribute__((ext_vector_type(2))) float v2f;
typedef __attribute__((ext_vector_type(8))) float v8f;

// ---------------------------------------------------------------------------
// fp32 GEMM via V_WMMA_F32_16X16X4_F32. One wave computes one 16x16 C tile.
// A: [M,K] row-major (K = KSTEPS*4, M multiple of 16 -> no A guards),
// B: [K,N] row-major, C: [M,N] row-major.
// ISA layout (CDNA5 ISA §7.12.2):
//   A 16x4 : lane holds row (lane&15); VGPR0/1 hold K = 2*(lane>>4) + {0,1}
//   B  4x16: lane holds col (lane&15); VGPR0/1 hold K = 2*(lane>>4) + {0,1}
//   C 16x16: VGPR i holds row i + 8*(lane>>4), col (lane&15)
// NGUARD only needed when N is not a multiple of 16 (layer 3, N=6).
// ---------------------------------------------------------------------------
template <int KSTEPS, bool NGUARD>
__global__ void gemm_f32_wmma(const float* __restrict__ A,
                              const float* __restrict__ B,
                              float* __restrict__ C,
                              int M, int N) {
  constexpr int K = KSTEPS * 4;
  const int wave = (int)((blockIdx.x * blockDim.x + threadIdx.x) >> 5);
  const int lane = (int)(threadIdx.x & 31);
  const int ntiles = (N + 15) >> 4;
  const int mtiles = M >> 4;
  if (wave >= ntiles * mtiles) return;  // uniform per-wave exit
  const int tm = wave / ntiles;
  const int tn = wave % ntiles;
  const int half = lane >> 4;           // 0 or 1
  const int l15  = lane & 15;
  const int row  = tm * 16 + l15;       // A row for this lane (always < M)
  const int col  = tn * 16 + l15;       // B / C column for this lane
  const int kh   = half * 2;

  const float* __restrict__ arow = A + (size_t)row * K + kh;  // 8B aligned
  const bool colok = !NGUARD || (col < N);

  v8f acc = {0.f, 0.f, 0.f, 0.f, 0.f, 0.f, 0.f, 0.f};
#pragma unroll
  for (int ks = 0; ks < KSTEPS; ++ks) {
    const int ka = ks * 4 + kh;
    const v2f a = *(const v2f*)(arow + ks * 4);   // global_load_b64
    v2f b;
    if (NGUARD) {
      b.x = colok ? B[(size_t)ka * N + col] : 0.f;
      b.y = colok ? B[(size_t)(ka + 1) * N + col] : 0.f;
    } else {
      b.x = B[(size_t)ka * N + col];
      b.y = B[(size_t)(ka + 1) * N + col];
    }
    acc = __builtin_amdgcn_wmma_f32_16x16x4_f32(false, a, false, b,
                                                (short)0, acc, false, false);
  }

  const int rbase = tm * 16 + half * 8;
  float* __restrict__ crow = C + (size_t)rbase * N + col;
#pragma unroll
  for (int i = 0; i < 8; ++i) {
    if (colok) crow[(size_t)i * N] = acc[i];
  }
}

// Zero-pad the K (inner) dimension: in [rows,K] -> out [rows,Kp]
__global__ void pad_inner(const float* __restrict__ in, float* __restrict__ out,
                          int rows, int K, int Kp) {
  const int idx = blockIdx.x * blockDim.x + threadIdx.x;
  if (idx >= rows * Kp) return;
  const int r = idx / Kp;
  const int k = idx % Kp;
  out[idx] = (k < K) ? in[r * K + k] : 0.f;
}

// Zero-pad the row (outer) dimension: in [inRows,cols] -> out [outRows,cols]
__global__ void pad_outer(const float* __restrict__ in, float* __restrict__ out,
                          int inRows, int outRows, int cols) {
  const int idx = blockIdx.x * blockDim.x + threadIdx.x;
  if (idx >= outRows * cols) return;
  const int r = idx / cols;
  out[idx] = (r < inRows) ? in[idx] : 0.f;
}

// el[n,h] = dot(feat[n,h,:], attn_l[h,:]);  er likewise.
__global__ void attn_proj(const float* __restrict__ feat,
                          const float* __restrict__ al,
                          const float* __restrict__ ar,
                          float* __restrict__ el, float* __restrict__ er,
                          int Nn, int Hh, int Fo) {
  const int idx = blockIdx.x * blockDim.x + threadIdx.x;
  if (idx >= Nn * Hh) return;
  const int n = idx / Hh;
  const int h = idx % Hh;
  const float* f = feat + (size_t)(n * Hh + h) * Fo;
  float sl = 0.f, sr = 0.f;
  for (int j = 0; j < Fo; ++j) {
    const float v = f[j];
    sl = fmaf(v, al[h * Fo + j], sl);
    sr = fmaf(v, ar[h * Fo + j], sr);
  }
  el[idx] = sl;
  er[idx] = sr;
}

// mx = -inf, sm = 0, rst = 0
__global__ void init_segments(float* __restrict__ mx, float* __restrict__ sm,
                              float* __restrict__ rst, int nh, int nrst) {
  const int i = blockIdx.x * blockDim.x + threadIdx.x;
  if (i < nh) { mx[i] = -INFINITY; sm[i] = 0.f; }
  if (i < nrst) rst[i] = 0.f;
}

__device__ __forceinline__ float atomicMaxFloat(float* addr, float val) {
  int* ia = (int*)addr;
  int old = __hip_atomic_load(ia, __ATOMIC_RELAXED, __HIP_MEMORY_SCOPE_AGENT);
  while (__int_as_float(old) < val) {
    const int assumed = old;
    old = atomicCAS(ia, assumed, __float_as_int(val));
    if (old == assumed) break;
  }
  return __int_as_float(old);
}

// e = leaky_relu(el[src]+er[dst]); segment max over dst.
__global__ void edge_logit_max(const float* __restrict__ el,
                               const float* __restrict__ er,
                               const int* __restrict__ src,
                               const int* __restrict__ dst,
                               float* __restrict__ ebuf, float* __restrict__ mx,
                               int E, int Hh) {
  const int idx = blockIdx.x * blockDim.x + threadIdx.x;
  if (idx >= E * Hh) return;
  const int e = idx / Hh;
  const int h = idx % Hh;
  const int sn = src[e], dn = dst[e];
  float v = el[sn * Hh + h] + er[dn * Hh + h];
  v = (v > 0.f) ? v : v * NEG_SLOPE;
  ebuf[idx] = v;
  atomicMaxFloat(&mx[dn * Hh + h], v);
}

// ex = exp(e - m[dst]); segment sum over dst.  (ebuf overwritten with ex)
__global__ void edge_exp_sum(float* __restrict__ ebuf,
                             const float* __restrict__ mx,
                             const int* __restrict__ dst,
                             float* __restrict__ sm, int E, int Hh) {
  const int idx = blockIdx.x * blockDim.x + threadIdx.x;
  if (idx >= E * Hh) return;
  const int e = idx / Hh;
  const int h = idx % Hh;
  const int dn = dst[e];
  const float ex = __expf(ebuf[idx] - mx[dn * Hh + h]);
  ebuf[idx] = ex;
  atomicAdd(&sm[dn * Hh + h], ex);
}

// rst[dst,h,:] += (ex/s[dst,h]) * feat[src,h,:]
__global__ void edge_aggregate(const float* __restrict__ ebuf,
                               const float* __restrict__ sm,
                               const float* __restrict__ feat,
                               const int* __restrict__ src,
                               const int* __restrict__ dst,
                               float* __restrict__ rst, int E, int Hh, int Fo) {
  const int idx = blockIdx.x * blockDim.x + threadIdx.x;
  if (idx >= E * Hh) return;
  const int e = idx / Hh;
  const int h = idx % Hh;
  const int sn = src[e], dn = dst[e];
  const float* fs = feat + (size_t)(sn * Hh + h) * Fo;
  __builtin_prefetch(fs, 0, 0);   // gfx1250 global_prefetch_b8: warm L2 for gather
  const float alpha = ebuf[idx] / sm[dn * Hh + h];
  float* rd = rst + (size_t)(dn * Hh + h) * Fo;
  for (int j = 0; j < Fo; ++j) atomicAdd(&rd[j], alpha * fs[j]);
}

// out[n,f] = sum_h (rst[n,h,f] + bias[h*Fo+f])
__global__ void head_sum_bias(const float* __restrict__ rst,
                              const float* __restrict__ bias,
                              float* __restrict__ out, int Nn, int Hh, int Fo) {
  const int idx = blockIdx.x * blockDim.x + threadIdx.x;
  if (idx >= Nn * Fo) return;
  const int n = idx / Fo;
  const int f = idx % Fo;
  float acc = 0.f;
  for (int h = 0; h < Hh; ++h)
    acc += rst[(size_t)(n * Hh + h) * Fo + f] + bias[h * Fo + f];
  out[idx] = acc;
}

// ---------------------------------------------------------------------------
// Everything after the GEMM is identical across layers; the GEMM itself is
// dispatched per-layer with the right template instantiation.
static void run_gat_tail(int Fo,
                         const float* al, const float* ar, const float* bias,
                         const int* src, const int* dst,
                         float* feat, float* rst, float* el, float* er,
                         float* mx, float* sm, float* ebuf,
                         float* out, hipStream_t stream) {
  const int Nout = HEADS * Fo;
  const int nh = N_NODES * HEADS;
  attn_proj<<<(nh + 255) / 256, 256, 0, stream>>>(feat, al, ar, el, er,
                                                  N_NODES, HEADS, Fo);
  const int nrst = N_NODES * Nout;
  const int ninit = (nh > nrst) ? nh : nrst;
  init_segments<<<(ninit + 255) / 256, 256, 0, stream>>>(mx, sm, rst, nh, nrst);
  const int eh = N_EDGES * HEADS;
  edge_logit_max<<<(eh + 255) / 256, 256, 0, stream>>>(el, er, src, dst, ebuf,
                                                       mx, N_EDGES, HEADS);
  edge_exp_sum<<<(eh + 255) / 256, 256, 0, stream>>>(ebuf, mx, dst, sm,
                                                     N_EDGES, HEADS);
  edge_aggregate<<<(eh + 255) / 256, 256, 0, stream>>>(ebuf, sm, feat, src, dst,
                                                       rst, N_EDGES, HEADS, Fo);
  const int nf = N_NODES * Fo;
  head_sum_bias<<<(nf + 255) / 256, 256, 0, stream>>>(rst, bias, out,
                                                      N_NODES, HEADS, Fo);
}

static inline int gemm_blocks(int N) {
  const int waves = (N_NODES / 16) * ((N + 15) / 16);
  return (waves * 32 + 255) / 256;
}

extern "C" void kernel_launch(void* const* d_in, const int* in_sizes, int n_in,
                              void* d_out, int out_size, void* d_ws, size_t ws_size,
                              hipStream_t stream) {
  const float* feats = (const float*)d_in[0];
  const int*   src   = (const int*)  d_in[1];
  const int*   dst   = (const int*)  d_in[2];
  const float* W1  = (const float*)d_in[3];
  const float* al1 = (const float*)d_in[4];
  const float* ar1 = (const float*)d_in[5];
  const float* b1  = (const float*)d_in[6];
  const float* W2  = (const float*)d_in[7];
  const float* al2 = (const float*)d_in[8];
  const float* ar2 = (const float*)d_in[9];
  const float* b2  = (const float*)d_in[10];
  const float* W3  = (const float*)d_in[11];
  const float* al3 = (const float*)d_in[12];
  const float* ar3 = (const float*)d_in[13];
  const float* b3  = (const float*)d_in[14];
  float* out = (float*)d_out;

  // Workspace carve-up (fp32, all sizes even -> every buffer 8B aligned).
  // Total ~26M floats (~104 MB) — resident in MI455X's 192 MB L2.
  float* ws      = (float*)d_ws;
  float* feat    = ws; ws += (size_t)N_NODES * HEADS * H_FEATS;   // [N, 192]
  float* rst     = ws; ws += (size_t)N_NODES * HEADS * H_FEATS;   // [N, 192]
  float* hbuf    = ws; ws += (size_t)N_NODES * H_FEATS;           // [N, 64]
  float* el      = ws; ws += (size_t)N_NODES * HEADS;
  float* er      = ws; ws += (size_t)N_NODES * HEADS;
  float* mx      = ws; ws += (size_t)N_NODES * HEADS;
  float* sm      = ws; ws += (size_t)N_NODES * HEADS;
  float* ebuf    = ws; ws += (size_t)N_EDGES * HEADS;
  float* featpad = ws; ws += (size_t)N_NODES * IN_FEATS_PAD;      // [N, 12]
  float* W1pad   = ws; ws += (size_t)IN_FEATS_PAD * HEADS * H_FEATS; // [12,192]

  const int N1 = HEADS * H_FEATS;    // 192
  const int N3 = HEADS * N_CLASSES;  // 6

  // ---- Layer 1: pad K 9->12, then unguarded WMMA GEMM (K=12, N=192) ----
  {
    const int np = N_NODES * IN_FEATS_PAD;
    pad_inner<<<(np + 255) / 256, 256, 0, stream>>>(feats, featpad,
                                                    N_NODES, IN_FEATS, IN_FEATS_PAD);
    const int wp = IN_FEATS_PAD * N1;
    pad_outer<<<(wp + 255) / 256, 256, 0, stream>>>(W1, W1pad,
                                                    IN_FEATS, IN_FEATS_PAD, N1);
    gemm_f32_wmma<IN_FEATS_PAD / 4, false>
        <<<gemm_blocks(N1), 256, 0, stream>>>(featpad, W1pad, feat, N_NODES, N1);
    run_gat_tail(H_FEATS, al1, ar1, b1, src, dst,
                 feat, rst, el, er, mx, sm, ebuf, hbuf, stream);
  }

  // ---- Layer 2: unguarded WMMA GEMM (K=64, N=192) ----
  {
    gemm_f32_wmma<H_FEATS / 4, false>
        <<<gemm_blocks(N1), 256, 0, stream>>>(hbuf, W2, feat, N_NODES, N1);
    run_gat_tail(H_FEATS, al2, ar2, b2, src, dst,
                 feat, rst, el, er, mx, sm, ebuf, hbuf, stream);
  }

  // ---- Layer 3: WMMA GEMM with N-guard (K=64, N=6) -> d_out [N,2] ----
  {
    gemm_f32_wmma<H_FEATS / 4, true>
        <<<gemm_blocks(N3), 256, 0, stream>>>(hbuf, W3, feat, N_NODES, N3);
    run_gat_tail(N_CLASSES, al3, ar3, b3, src, dst,
                 feat, rst, el, er, mx, sm, ebuf, out, stream);
  }
}